// GraphNetwork_31172872634708
// MI455X (gfx1250) — compile-verified
//
#include <hip/hip_runtime.h>
#include <hip/hip_bf16.h>

#define F_  256
#define H_  128
#define R_  16
#define NB_ 30
#define C_  7

#define KC      64      // K-chunk staged in LDS
#define LDA_LDS 68      // padded A row stride (floats): 16B-aligned rows, conflict-free
#define LDB_LDS 136     // padded B row stride (floats): lane-halves hit disjoint banks

typedef float v2f __attribute__((ext_vector_type(2)));
typedef float v8f __attribute__((ext_vector_type(8)));
typedef int   vi4 __attribute__((vector_size(16)));   // matches builtin's b128 pointee

// ---------------------------------------------------------------------------
// fp32 WMMA 16x16x4 (CDNA5): D = A(16x4) * B(4x16) + C(16x16 f32)
// ---------------------------------------------------------------------------
__device__ __forceinline__ v8f wmma4(v2f a, v2f b, v8f c) {
  return __builtin_amdgcn_wmma_f32_16x16x4_f32(
      /*neg_a=*/false, a, /*neg_b=*/false, b,
      /*c_mod=*/(short)0, c, /*reuse_a=*/false, /*reuse_b=*/false);
}

// ---------------------------------------------------------------------------
// 16-byte global -> LDS copy. Prefer CDNA5 async-to-LDS DMA; fall back to a
// regular b128 load + ds_store if the builtin is unavailable.
// ---------------------------------------------------------------------------
#if defined(__gfx1250__) && __has_builtin(__builtin_amdgcn_global_load_async_to_lds_b128)
#define ASYNC_LDS_COPY 1
#else
#define ASYNC_LDS_COPY 0
#endif

__device__ __forceinline__ void copy16_to_lds(const float* __restrict__ g,
                                              float* __restrict__ l) {
#if ASYNC_LDS_COPY
  __builtin_amdgcn_global_load_async_to_lds_b128(
      (__attribute__((address_space(1))) vi4*)(g),
      (__attribute__((address_space(3))) vi4*)(l),
      /*offset=*/0, /*cpol=*/0);
#else
  float4 v = *(const float4*)g;
  *(float4*)l = v;
#endif
}

__device__ __forceinline__ void stage_join() {
#if ASYNC_LDS_COPY
  asm volatile("s_wait_asynccnt 0" ::: "memory");
#endif
  __syncthreads();
}

// ---------------------------------------------------------------------------
// Block GEMM: 256 threads, block tile 64(M) x 128(N), K-chunks of 64 staged in
// LDS. Waves arranged 4(M) x 2(N); each wave owns a 16x64 strip = 4 WMMA tiles.
// A row-major (lda), B row-major with ldb == 128. Accumulates into c[4].
// ---------------------------------------------------------------------------
__device__ __forceinline__ void block_gemm_acc(const float* __restrict__ A, int lda,
                                               const float* __restrict__ B,
                                               int row0, int K, int Nn,
                                               float* __restrict__ Alds,
                                               float* __restrict__ Blds,
                                               v8f c[4], bool live) {
  const int t     = threadIdx.x;
  const int lane  = t & 31;
  const int wave  = t >> 5;
  const int half  = lane >> 4;
  const int l16   = lane & 15;
  const int mrow  = (wave & 3) * 16;   // wave tile row within block
  const int nbase = (wave >> 2) * 64;  // wave strip base column

  for (int k = 0; k < K; k += KC) {
    // ---- stage A chunk: 64 rows x 64 cols = 1024 x b128 (4 per thread) ----
#pragma unroll
    for (int i = 0; i < 4; ++i) {
      int s   = t + i * 256;
      int row = s >> 4;
      int seg = (s & 15) * 4;
      if (row0 + row < Nn)
        copy16_to_lds(A + (size_t)(row0 + row) * lda + k + seg,
                      Alds + row * LDA_LDS + seg);
    }
    // ---- stage B chunk: 64 rows x 128 cols = 2048 x b128 (8 per thread) ----
#pragma unroll
    for (int i = 0; i < 8; ++i) {
      int s   = t + i * 256;
      int row = s >> 5;
      int seg = (s & 31) * 4;
      copy16_to_lds(B + (size_t)(k + row) * H_ + seg,
                    Blds + row * LDB_LDS + seg);
    }
    stage_join();

    if (live) {
#pragma unroll 4
      for (int kk = 0; kk < KC; kk += 4) {
        const int ka = kk + 2 * half;
        v2f a;
        a.x = Alds[(mrow + l16) * LDA_LDS + ka];
        a.y = Alds[(mrow + l16) * LDA_LDS + ka + 1];
#pragma unroll
        for (int tt = 0; tt < 4; ++tt) {
          v2f b;
          b.x = Blds[ka * LDB_LDS + nbase + tt * 16 + l16];
          b.y = Blds[(ka + 1) * LDB_LDS + nbase + tt * 16 + l16];
          c[tt] = wmma4(a, b, c[tt]);
        }
      }
    }
    __syncthreads();   // protect LDS before next chunk overwrites it
  }
}

// 16x16 f32 C/D tile <-> memory (VGPR v: M = v + 8*half, N = l16), ldd = 128
__device__ __forceinline__ v8f load_tile(const float* __restrict__ D,
                                         int row0, int col0) {
  const int lane = threadIdx.x & 31;
  const int half = lane >> 4;
  const int l16  = lane & 15;
  v8f c;
#pragma unroll
  for (int v = 0; v < 8; ++v)
    c[v] = D[(size_t)(row0 + v + 8 * half) * H_ + col0 + l16];
  return c;
}

__device__ __forceinline__ void store_tile(float* __restrict__ D,
                                           int row0, int col0, const v8f& c) {
  const int lane = threadIdx.x & 31;
  const int half = lane >> 4;
  const int l16  = lane & 15;
#pragma unroll
  for (int v = 0; v < 8; ++v)
    D[(size_t)(row0 + v + 8 * half) * H_ + col0 + l16] = c[v];
}

// Per-wave tile coordinates within the 64x128 block tile.
__device__ __forceinline__ void wave_coords(int Nn, int& row0, int& nbase, bool& live) {
  const int wave = threadIdx.x >> 5;
  row0  = blockIdx.x * 64 + (wave & 3) * 16;
  nbase = (wave >> 2) * 64;
  live  = (row0 < Nn);    // wave-uniform; Nn % 16 == 0
}

// ---------------------------------------------------------------------------
// Kernels
// ---------------------------------------------------------------------------

// W[r,i,o] = sum_b comp[r,b] * basis[b,i,o]
__global__ void compute_w_kernel(const float* __restrict__ comp,
                                 const float* __restrict__ basis,
                                 float* __restrict__ W, int total) {
  int idx = blockIdx.x * blockDim.x + threadIdx.x;
  if (idx >= total) return;
  int r  = idx / (F_ * H_);
  int io = idx - r * (F_ * H_);
  float acc = 0.f;
#pragma unroll
  for (int b = 0; b < NB_; ++b)
    acc = fmaf(comp[r * NB_ + b], basis[(size_t)b * F_ * H_ + io], acc);
  W[idx] = acc;
}

// cnt[dst*R + type] += 1
__global__ void count_kernel(const int* __restrict__ dst,
                             const int* __restrict__ etype,
                             float* __restrict__ cnt, int E) {
  int e = blockIdx.x * blockDim.x + threadIdx.x;
  if (e >= E) return;
  atomicAdd(&cnt[(size_t)dst[e] * R_ + etype[e]], 1.0f);
}

// inv = 1 / max(cnt, 1)   (in place)
__global__ void inv_kernel(float* __restrict__ cnt, int total) {
  int i = blockIdx.x * blockDim.x + threadIdx.x;
  if (i >= total) return;
  cnt[i] = 1.0f / fmaxf(cnt[i], 1.0f);
}

// xWr = x @ W[r]        [Nn x 256] @ [256 x 128]
__global__ void __launch_bounds__(256)
gemm_xw_kernel(const float* __restrict__ x, const float* __restrict__ Wr,
               float* __restrict__ xWr, int Nn) {
  __shared__ float Alds[64 * LDA_LDS];
  __shared__ float Blds[KC * LDB_LDS];
  int row0, nbase; bool live;
  wave_coords(Nn, row0, nbase, live);
  v8f c[4] = {};
  block_gemm_acc(x, F_, Wr, blockIdx.x * 64, F_, Nn, Alds, Blds, c, live);
  if (live) {
#pragma unroll
    for (int tt = 0; tt < 4; ++tt) store_tile(xWr, row0, nbase + tt * 16, c[tt]);
  }
}

// out1[dst] += inv[dst*R+r] * xWr[src]   for edges of relation r (warp per edge)
__global__ void scatter_rel_kernel(const int* __restrict__ src,
                                   const int* __restrict__ dst,
                                   const int* __restrict__ etype,
                                   const float* __restrict__ inv,
                                   const float* __restrict__ xWr,
                                   float* __restrict__ out1, int E, int r) {
  const int lane = threadIdx.x & 31;
  int gw = (blockIdx.x * blockDim.x + threadIdx.x) >> 5;
  int nw = (gridDim.x * blockDim.x) >> 5;
  for (int e = gw; e < E; e += nw) {
    if (etype[e] != r) continue;
    int s = src[e], d = dst[e];
    float w = inv[(size_t)d * R_ + r];
    float4 v = ((const float4*)(xWr + (size_t)s * H_))[lane];
    float* o = out1 + (size_t)d * H_ + lane * 4;
    atomicAdd(o + 0, w * v.x);
    atomicAdd(o + 1, w * v.y);
    atomicAdd(o + 2, w * v.z);
    atomicAdd(o + 3, w * v.w);
  }
}

// out1 += x @ root1 + bias1   (accumulator initialized from out1 = agg1)
__global__ void __launch_bounds__(256)
gemm_root_kernel(const float* __restrict__ x, const float* __restrict__ root1,
                 const float* __restrict__ bias1, float* __restrict__ out1, int Nn) {
  __shared__ float Alds[64 * LDA_LDS];
  __shared__ float Blds[KC * LDB_LDS];
  int row0, nbase; bool live;
  wave_coords(Nn, row0, nbase, live);
  v8f c[4] = {};
  if (live) {
#pragma unroll
    for (int tt = 0; tt < 4; ++tt) c[tt] = load_tile(out1, row0, nbase + tt * 16);
  }
  block_gemm_acc(x, F_, root1, blockIdx.x * 64, F_, Nn, Alds, Blds, c, live);
  if (live) {
#pragma unroll
    for (int tt = 0; tt < 4; ++tt) {
      const float b = bias1[nbase + tt * 16 + (threadIdx.x & 15)];
      v8f cc = c[tt];
#pragma unroll
      for (int v = 0; v < 8; ++v) cc[v] += b;
      store_tile(out1, row0, nbase + tt * 16, cc);
    }
  }
}

// agg2[dst] += out1[src]   (warp per edge, all edges)
__global__ void scatter_all_kernel(const int* __restrict__ src,
                                   const int* __restrict__ dst,
                                   const float* __restrict__ out1,
                                   float* __restrict__ agg2, int E) {
  const int lane = threadIdx.x & 31;
  int gw = (blockIdx.x * blockDim.x + threadIdx.x) >> 5;
  int nw = (gridDim.x * blockDim.x) >> 5;
  for (int e = gw; e < E; e += nw) {
    int s = src[e], d = dst[e];
    float4 v = ((const float4*)(out1 + (size_t)s * H_))[lane];
    float* o = agg2 + (size_t)d * H_ + lane * 4;
    atomicAdd(o + 0, v.x);
    atomicAdd(o + 1, v.y);
    atomicAdd(o + 2, v.z);
    atomicAdd(o + 3, v.w);
  }
}

// out2 = agg2 @ w_rel + out1 @ w_root + bias2
__global__ void __launch_bounds__(256)
gemm_out2_kernel(const float* __restrict__ agg2, const float* __restrict__ w_rel,
                 const float* __restrict__ out1, const float* __restrict__ w_root,
                 const float* __restrict__ bias2, float* __restrict__ out2, int Nn) {
  __shared__ float Alds[64 * LDA_LDS];
  __shared__ float Blds[KC * LDB_LDS];
  int row0, nbase; bool live;
  wave_coords(Nn, row0, nbase, live);
  v8f c[4] = {};
  block_gemm_acc(agg2, H_, w_rel,  blockIdx.x * 64, H_, Nn, Alds, Blds, c, live);
  block_gemm_acc(out1, H_, w_root, blockIdx.x * 64, H_, Nn, Alds, Blds, c, live);
  if (live) {
#pragma unroll
    for (int tt = 0; tt < 4; ++tt) {
      const float b = bias2[nbase + tt * 16 + (threadIdx.x & 15)];
      v8f cc = c[tt];
#pragma unroll
      for (int v = 0; v < 8; ++v) cc[v] += b;
      store_tile(out2, row0, nbase + tt * 16, cc);
    }
  }
}

// hidden = relu(concat(x, out2) @ lin_w + lin_b)
__global__ void __launch_bounds__(256)
gemm_hidden_kernel(const float* __restrict__ x, const float* __restrict__ out2,
                   const float* __restrict__ lin_w, const float* __restrict__ lin_b,
                   float* __restrict__ hid, int Nn) {
  __shared__ float Alds[64 * LDA_LDS];
  __shared__ float Blds[KC * LDB_LDS];
  int row0, nbase; bool live;
  wave_coords(Nn, row0, nbase, live);
  v8f c[4] = {};
  block_gemm_acc(x,    F_, lin_w,                   blockIdx.x * 64, F_, Nn, Alds, Blds, c, live);
  block_gemm_acc(out2, H_, lin_w + (size_t)F_ * H_, blockIdx.x * 64, H_, Nn, Alds, Blds, c, live);
  if (live) {
#pragma unroll
    for (int tt = 0; tt < 4; ++tt) {
      const float b = lin_b[nbase + tt * 16 + (threadIdx.x & 15)];
      v8f cc = c[tt];
#pragma unroll
      for (int v = 0; v < 8; ++v) cc[v] = fmaxf(cc[v] + b, 0.0f);
      store_tile(hid, row0, nbase + tt * 16, cc);
    }
  }
}

// logits = hidden @ fc_w + fc_b; out = log_softmax(logits)  (thread per node)
__global__ void head_kernel(const float* __restrict__ hid,
                            const float* __restrict__ fc_w,
                            const float* __restrict__ fc_b,
                            float* __restrict__ out, int Nn) {
  int n = blockIdx.x * blockDim.x + threadIdx.x;
  if (n >= Nn) return;
  float acc[C_];
#pragma unroll
  for (int c = 0; c < C_; ++c) acc[c] = fc_b[c];
  const float* h = hid + (size_t)n * H_;
  for (int k = 0; k < H_; ++k) {
    float hv = h[k];
#pragma unroll
    for (int c = 0; c < C_; ++c) acc[c] = fmaf(hv, fc_w[k * C_ + c], acc[c]);
  }
  float m = acc[0];
#pragma unroll
  for (int c = 1; c < C_; ++c) m = fmaxf(m, acc[c]);
  float s = 0.f;
#pragma unroll
  for (int c = 0; c < C_; ++c) s += __expf(acc[c] - m);
  float lse = m + __logf(s);
#pragma unroll
  for (int c = 0; c < C_; ++c) out[(size_t)n * C_ + c] = acc[c] - lse;
}

// ---------------------------------------------------------------------------
// Launch
// ---------------------------------------------------------------------------
extern "C" void kernel_launch(void* const* d_in, const int* in_sizes, int n_in,
                              void* d_out, int out_size, void* d_ws, size_t ws_size,
                              hipStream_t stream) {
  const float* x          = (const float*)d_in[0];
  const int*   edge_index = (const int*)d_in[1];
  const int*   etype      = (const int*)d_in[3];
  const float* basis      = (const float*)d_in[8];
  const float* comp       = (const float*)d_in[9];
  const float* root1      = (const float*)d_in[10];
  const float* bias1      = (const float*)d_in[11];
  const float* w_rel      = (const float*)d_in[12];
  const float* w_root     = (const float*)d_in[13];
  const float* bias2      = (const float*)d_in[14];
  const float* lin_w      = (const float*)d_in[15];
  const float* lin_b      = (const float*)d_in[16];
  const float* fc_w       = (const float*)d_in[17];
  const float* fc_b       = (const float*)d_in[18];
  float* out = (float*)d_out;

  const int Nn = in_sizes[0] / F_;    // 50000
  const int E  = in_sizes[1] / 2;     // 800000
  const int* src = edge_index;
  const int* dst = edge_index + E;

  // workspace carve-out (all offsets 16B aligned): ~127 MB total
  float* ws   = (float*)d_ws;
  float* W    = ws;                           // R*F*H
  float* inv  = W    + (size_t)R_ * F_ * H_;  // Nn*R (counts, then inverted)
  float* xWr  = inv  + (size_t)Nn * R_;       // Nn*H (per-relation GEMM output)
  float* out1 = xWr  + (size_t)Nn * H_;       // Nn*H
  float* agg2 = out1 + (size_t)Nn * H_;       // Nn*H
  float* out2 = agg2 + (size_t)Nn * H_;       // Nn*H
  float* hid  = out2 + (size_t)Nn * H_;       // Nn*H

  (void)hipMemsetAsync(inv,  0, (size_t)Nn * R_ * sizeof(float), stream);
  (void)hipMemsetAsync(out1, 0, (size_t)Nn * H_ * sizeof(float), stream);
  (void)hipMemsetAsync(agg2, 0, (size_t)Nn * H_ * sizeof(float), stream);

  { // relation weight matrices
    int total = R_ * F_ * H_;
    compute_w_kernel<<<(total + 255) / 256, 256, 0, stream>>>(comp, basis, W, total);
  }
  count_kernel<<<(E + 255) / 256, 256, 0, stream>>>(dst, etype, inv, E);
  inv_kernel<<<(Nn * R_ + 255) / 256, 256, 0, stream>>>(inv, Nn * R_);

  dim3 blk(256);
  dim3 grd((Nn + 63) / 64);           // 782 blocks, each covers 64 rows x 128 cols
  for (int r = 0; r < R_; ++r) {
    gemm_xw_kernel<<<grd, blk, 0, stream>>>(x, W + (size_t)r * F_ * H_, xWr, Nn);
    scatter_rel_kernel<<<1024, 256, 0, stream>>>(src, dst, etype, inv, xWr, out1, E, r);
  }
  gemm_root_kernel<<<grd, blk, 0, stream>>>(x, root1, bias1, out1, Nn);
  scatter_all_kernel<<<2048, 256, 0, stream>>>(src, dst, out1, agg2, E);
  gemm_out2_kernel<<<grd, blk, 0, stream>>>(agg2, w_rel, out1, w_root, bias2, out2, Nn);
  gemm_hidden_kernel<<<grd, blk, 0, stream>>>(x, out2, lin_w, lin_b, hid, Nn);
  head_kernel<<<(Nn + 255) / 256, 256, 0, stream>>>(hid, fc_w, fc_b, out, Nn);
}